// LightGCN_19576460935173
// MI455X (gfx1250) — compile-verified
//
#include <hip/hip_runtime.h>

#define LATENT 64
#define SPMM_BLOCK 128
#define SPMM_WAVES (SPMM_BLOCK / 32)
#define SPMM_GRID 8192

// ---------------------------------------------------------------------------
// Kernel 1: emb = hyp_logmap0(hyp_proj(w, 1), 1), one wave32 per row.
// y = w[1:64]; s = sum(y^2); x0 = sqrt(1+s);
// out[0] = 0; out[1:] = arcosh(max(x0,1+eps)) / max(sqrt(s),1e-15) * y
// ---------------------------------------------------------------------------
__global__ void hyp_logmap_kernel(const float* __restrict__ user_w,
                                  const float* __restrict__ item_w,
                                  float* __restrict__ emb,
                                  int n_users, int n_total) {
  const int gwave = (int)((blockIdx.x * blockDim.x + threadIdx.x) >> 5);
  const int lane = threadIdx.x & 31;
  if (gwave >= n_total) return;

  const float* src = (gwave < n_users)
      ? user_w + (size_t)gwave * LATENT
      : item_w + (size_t)(gwave - n_users) * LATENT;

  float2 v = ((const float2*)src)[lane];          // lane l holds w[2l], w[2l+1]
  float ax = (lane == 0) ? 0.0f : v.x;            // exclude w[0] from y
  float s = ax * ax + v.y * v.y;
  #pragma unroll
  for (int m = 16; m > 0; m >>= 1) s += __shfl_xor(s, m, 32);  // wave32 reduce

  float x0    = sqrtf(fmaxf(1.0f + s, 1e-7f));
  float theta = fmaxf(x0, 1.0f + 1e-7f);
  float arc   = logf(theta + sqrtf(theta * theta - 1.0f));     // arcosh
  float ynorm = fmaxf(sqrtf(s), 1e-15f);
  float scale = arc / ynorm;

  float2 o;
  o.x = (lane == 0) ? 0.0f : v.x * scale;
  o.y = v.y * scale;
  ((float2*)(emb + (size_t)gwave * LATENT))[lane] = o;
}

// ---------------------------------------------------------------------------
// Kernel 2: scatter SpMM  emb_out[rows[e]] += vals[e] * emb_in[cols[e]]
// Each wave owns a CONTIGUOUS chunk of edge pairs (streams the 12B/edge index
// arrays cacheline-by-cacheline). Per iteration a wave handles 2 nnz
// (16 lanes x float4 per 256B row). Row gathers are staged global->LDS with
// the gfx1250 async path (global_load_async_to_lds_b128, ASYNCcnt-tracked),
// double-buffered: next gather + next edge indices are in flight while the
// current row's global_atomic_add_f32 stream drains into L2.
// ---------------------------------------------------------------------------
__global__ void spmm_scatter_kernel(const int* __restrict__ rows,
                                    const int* __restrict__ cols,
                                    const float* __restrict__ vals,
                                    const float* __restrict__ emb_in,
                                    float* __restrict__ emb_out,
                                    int nnz) {
  __shared__ __align__(16) float smem[SPMM_WAVES][2][2 * LATENT];

  const int lane   = threadIdx.x & 31;
  const int wloc   = threadIdx.x >> 5;
  const int wave   = blockIdx.x * SPMM_WAVES + wloc;
  const int nwaves = gridDim.x * SPMM_WAVES;
  const int half   = lane >> 4;   // which of the pair's 2 nnz
  const int q      = lane & 15;   // which float4 of the 64-float row

  const int npairs = (nnz + 1) >> 1;
  const int chunk  = (npairs + nwaves - 1) / nwaves;   // contiguous pairs/wave
  const int pbeg   = wave * chunk;
  const int pend   = min(pbeg + chunk, npairs);
  if (pbeg >= pend) return;
  const int niter = pend - pbeg;

  float* lbase0 = &smem[wloc][0][half * LATENT + q * 4];
  float* lbase1 = &smem[wloc][1][half * LATENT + q * 4];

  auto stage = [&](float* lptr, int e) {
    int ec = (e < nnz) ? e : (nnz - 1);           // clamp: keep EXEC all-ones
    const float* gp = emb_in + (size_t)cols[ec] * LATENT + q * 4;
    // low 32 bits of the flat shared pointer == LDS byte offset
    unsigned lds_off = (unsigned)(unsigned long long)(uintptr_t)lptr;
    asm volatile("global_load_async_to_lds_b128 %0, %1, off"
                 :: "v"(lds_off), "v"(gp) : "memory");
  };

  stage(lbase0, 2 * pbeg + half);                  // prologue
  int slot = 0;
  for (int it = 0; it < niter; ++it) {
    const int  ecur  = 2 * (pbeg + it) + half;
    const bool valid = ecur < nnz;
    const int  eC    = valid ? ecur : (nnz - 1);
    const int   r = rows[eC];                      // issue before the wait:
    const float v = vals[eC];                      // overlaps async latency
    if (it + 1 < niter) {
      stage(slot ? lbase0 : lbase1, 2 * (pbeg + it + 1) + half);
      asm volatile("s_wait_asynccnt 0x1" ::: "memory");  // current slot done
    } else {
      asm volatile("s_wait_asynccnt 0x0" ::: "memory");
    }
    float4 d = *(const float4*)(slot ? lbase1 : lbase0);
    if (valid) {
      float* op = emb_out + (size_t)r * LATENT + q * 4;
      unsafeAtomicAdd(op + 0, v * d.x);            // global_atomic_add_f32
      unsafeAtomicAdd(op + 1, v * d.y);
      unsafeAtomicAdd(op + 2, v * d.z);
      unsafeAtomicAdd(op + 3, v * d.w);
    }
    slot ^= 1;
  }
}

// ---------------------------------------------------------------------------
// Kernel 3: out = concat[ light[user], light[NUM_USERS+pos], light[items] ]
// with light = B0 + B1 + B2 computed on the fly. One float4 per thread.
// ---------------------------------------------------------------------------
__global__ void gather_out_kernel(const float* __restrict__ b0,
                                  const float* __restrict__ b1,
                                  const float* __restrict__ b2,
                                  const int* __restrict__ user,
                                  const int* __restrict__ pos,
                                  float* __restrict__ out,
                                  int batch, int n_users, int total4) {
  int idx = blockIdx.x * blockDim.x + threadIdx.x;
  if (idx >= total4) return;
  int o = idx >> 4;            // output row (16 float4 per 64-float row)
  int j = idx & 15;            // float4 within row
  int r;
  if (o < batch)            r = user[o];
  else if (o < 2 * batch)   r = n_users + pos[o - batch];
  else                      r = n_users + (o - 2 * batch);
  size_t off = (size_t)r * LATENT + j * 4;
  float4 a = *(const float4*)(b0 + off);
  float4 b = *(const float4*)(b1 + off);
  float4 c = *(const float4*)(b2 + off);
  float4 s;
  s.x = a.x + b.x + c.x;
  s.y = a.y + b.y + c.y;
  s.z = a.z + b.z + c.z;
  s.w = a.w + b.w + c.w;
  ((float4*)out)[idx] = s;
}

// ---------------------------------------------------------------------------
extern "C" void kernel_launch(void* const* d_in, const int* in_sizes, int n_in,
                              void* d_out, int out_size, void* d_ws, size_t ws_size,
                              hipStream_t stream) {
  const float* user_w    = (const float*)d_in[0];
  const float* item_w    = (const float*)d_in[1];
  const float* edge_vals = (const float*)d_in[2];
  const int*   edge_rows = (const int*)d_in[3];
  const int*   edge_cols = (const int*)d_in[4];
  const int*   user      = (const int*)d_in[5];
  const int*   pos       = (const int*)d_in[6];

  const int n_users = in_sizes[0] / LATENT;
  const int n_items = in_sizes[1] / LATENT;
  const int n_total = n_users + n_items;
  const int nnz     = in_sizes[2];
  const int batch   = in_sizes[5];

  const size_t bufsz = (size_t)n_total * LATENT * sizeof(float);
  float* B0 = (float*)d_ws;                        // emb0, later A^3 e0
  float* B1 = (float*)((char*)d_ws + bufsz);       // A e0
  float* B2 = (float*)((char*)d_ws + 2 * bufsz);   // A^2 e0

  // 1) initial hyperbolic logmap embedding -> B0
  {
    const int waves_per_block = 256 / 32;
    const int grid = (n_total + waves_per_block - 1) / waves_per_block;
    hyp_logmap_kernel<<<grid, 256, 0, stream>>>(user_w, item_w, B0,
                                                n_users, n_total);
  }

  // 2) layer 1: B1 = A * B0
  hipMemsetAsync(B1, 0, bufsz, stream);
  spmm_scatter_kernel<<<SPMM_GRID, SPMM_BLOCK, 0, stream>>>(
      edge_rows, edge_cols, edge_vals, B0, B1, nnz);

  // 3) layer 2: B2 = A * B1
  hipMemsetAsync(B2, 0, bufsz, stream);
  spmm_scatter_kernel<<<SPMM_GRID, SPMM_BLOCK, 0, stream>>>(
      edge_rows, edge_cols, edge_vals, B1, B2, nnz);

  // 4) layer 3: B0 = A * B2   (B0's original content no longer needed)
  hipMemsetAsync(B0, 0, bufsz, stream);
  spmm_scatter_kernel<<<SPMM_GRID, SPMM_BLOCK, 0, stream>>>(
      edge_rows, edge_cols, edge_vals, B2, B0, nnz);

  // 5) gather output: light = B0 + B1 + B2 at [user, n_users+pos, all items]
  {
    const int total4 = out_size / 4;
    const int grid = (total4 + 255) / 256;
    gather_out_kernel<<<grid, 256, 0, stream>>>(B0, B1, B2, user, pos,
                                                (float*)d_out,
                                                batch, n_users, total4);
  }
}